// ScalingLayer_28329604284563
// MI455X (gfx1250) — compile-verified
//
#include <hip/hip_runtime.h>
#include <stdint.h>

// ---------------------------------------------------------------------------
// ScalingLayer: y = x * exp(S) (column-wise), det_out = det_in + sum(S)
// Pure streaming kernel (AI ~0.13 flop/byte -> HBM-roof bound, ~22us at
// 23.3 TB/s). Hot loop = NT b128 load -> 2x v_pk_mul_f32 -> NT b128 store.
// Per-thread column scale held in registers (grid stride is a multiple of
// D/4, so each thread's column group is loop-invariant).
// ---------------------------------------------------------------------------

typedef __attribute__((ext_vector_type(4))) float f32x4;

__global__ __launch_bounds__(256) void scale_cols_nt_kernel(
    const float* __restrict__ x, const float* __restrict__ S,
    float* __restrict__ y, int d4, long total4) {
  const f32x4* __restrict__ x4 = (const f32x4*)x;
  const f32x4* __restrict__ S4 = (const f32x4*)S;
  f32x4* __restrict__ y4 = (f32x4*)y;

  const long stride = (long)gridDim.x * (long)blockDim.x;  // multiple of d4
  const int mask = d4 - 1;  // d4 is a power of two (256 for D=1024)

  const long i0 = (long)blockIdx.x * blockDim.x + threadIdx.x;

  // This thread's 4 columns are fixed for the whole loop (stride % d4 == 0):
  // compute exp(S) for them once, keep in VGPRs. S is 4 KB -> L2 resident.
  f32x4 sv = S4[(int)i0 & mask];
  f32x4 ev;
  ev.x = __expf(sv.x);
  ev.y = __expf(sv.y);
  ev.z = __expf(sv.z);
  ev.w = __expf(sv.w);

  const long stride2 = stride * 2;
  long i = i0;

  // 2x unrolled streaming loop: two independent NT b128 loads/stores per
  // iteration form memory clauses; prefetch one pair ahead
  // (global_prefetch_b8; speculative, silently dropped past end of buffer).
  for (; i + stride < total4; i += stride2) {
    __builtin_prefetch(&x4[i + stride2], 0, 0);
    __builtin_prefetch(&x4[i + stride2 + stride], 0, 0);
    f32x4 a = __builtin_nontemporal_load(&x4[i]);
    f32x4 b = __builtin_nontemporal_load(&x4[i + stride]);
    f32x4 ra, rb;
    ra.x = a.x * ev.x; ra.y = a.y * ev.y; ra.z = a.z * ev.z; ra.w = a.w * ev.w;
    rb.x = b.x * ev.x; rb.y = b.y * ev.y; rb.z = b.z * ev.z; rb.w = b.w * ev.w;
    __builtin_nontemporal_store(ra, &y4[i]);
    __builtin_nontemporal_store(rb, &y4[i + stride]);
  }
  if (i < total4) {
    f32x4 a = __builtin_nontemporal_load(&x4[i]);
    f32x4 ra;
    ra.x = a.x * ev.x; ra.y = a.y * ev.y; ra.z = a.z * ev.z; ra.w = a.w * ev.w;
    __builtin_nontemporal_store(ra, &y4[i]);
  }
}

__global__ __launch_bounds__(256) void det_add_kernel(
    const float* __restrict__ det_in, const float* __restrict__ S,
    float* __restrict__ det_out, int D, int B) {
  __shared__ float red[256];
  const int t = threadIdx.x;

  // Each block redundantly reduces S (1024 floats, L2-resident -> negligible).
  float s = 0.0f;
  for (int j = t; j < D; j += 256) s += S[j];
  red[t] = s;
  __syncthreads();
  for (int off = 128; off > 0; off >>= 1) {
    if (t < off) red[t] += red[t + off];
    __syncthreads();
  }
  const float logdet = red[0];

  const int idx = blockIdx.x * 256 + t;
  if (idx < B) det_out[idx] = det_in[idx] + logdet;
}

extern "C" void kernel_launch(void* const* d_in, const int* in_sizes, int n_in,
                              void* d_out, int out_size, void* d_ws, size_t ws_size,
                              hipStream_t stream) {
  (void)n_in; (void)out_size; (void)d_ws; (void)ws_size;

  const float* x      = (const float*)d_in[0];  // (B, D) f32
  const float* det_in = (const float*)d_in[1];  // (B,)   f32
  const float* S      = (const float*)d_in[2];  // (D,)   f32

  const long total = (long)in_sizes[0];   // B*D
  const int  B     = in_sizes[1];
  const int  D     = in_sizes[2];
  const long total4 = total / 4;
  const int  d4     = D / 4;              // power of two (256 for D=1024)

  float* y       = (float*)d_out;          // first output: (B, D)
  float* det_out = (float*)d_out + total;  // second output: (B,)

  // 4096 blocks x 8 wave32 = 32768 waves; stride = 4096*256 is a multiple of
  // d4, keeping each thread's column group loop-invariant.
  int blocks = 4096;
  long need = (total4 + 255) / 256;
  if ((long)blocks > need) blocks = (int)need;
  if (blocks < 1) blocks = 1;

  scale_cols_nt_kernel<<<blocks, 256, 0, stream>>>(x, S, y, d4, total4);
  det_add_kernel<<<(B + 255) / 256, 256, 0, stream>>>(det_in, S, det_out, D, B);
}